// ScalEdgesRead_22651657519351
// MI455X (gfx1250) — compile-verified
//
#include <hip/hip_runtime.h>

// ---------------------------------------------------------------------------
// ScalEdgesRead for MI455X (gfx1250):
//   Phase 1 (WMMA): h[which][row][dout] = sum_k x[row][k] * W_which[dout][k]
//                   for all B*N node rows, which in {src,dst}.  h lives in
//                   d_ws (51.2 MB -> resident in 192 MB L2).
//   Phase 2:        out[b][e][:] = (h[0][b*N+src[e]][:] + h[1][b*N+dst[e]][:])
//                   * 1/sqrt(2), 128-bit vectorized, non-temporal output
//                   stores so the 409.6 MB stream does not evict h from L2.
// ---------------------------------------------------------------------------

typedef __attribute__((ext_vector_type(2))) float v2f;
typedef __attribute__((ext_vector_type(4))) float v4f;
typedef __attribute__((ext_vector_type(8))) float v8f;

#define INV_SQRT_2 0.70710678118654752440f

// ---------------------------------------------------------------------------
// Phase 1: node transform via V_WMMA_F32_16X16X4_F32.
// One wave = one 16-row tile x one 16-col tile x one weight matrix.
// Block of 256 threads = 8 waves = all (which, otile) combos for one row tile.
//
// A-matrix 16x4 f32 layout (ISA 7.12.2): lanes 0-15 hold M=lane, K=4kk+{0,1};
// lanes 16-31 hold M=lane-16, K=4kk+{2,3}  -> contiguous float2 loads.
// B-matrix 4x16: same pattern with N across lanes; B[k][n] = W[n][k], so the
// per-lane B fragment is W[n_tile*16 + (lane&15)][k..k+1] -> float2 loads.
// C/D 16x16 f32: VGPR r = row (r or r+8), col = lane&15.
// ---------------------------------------------------------------------------
__global__ void node_transform_wmma(const float* __restrict__ x,     // [BN][64]
                                    const float* __restrict__ Wsrc,  // [64][64]
                                    const float* __restrict__ Wdst,  // [64][64]
                                    float* __restrict__ h,           // [2][BN][64]
                                    int BN) {
  const int lane  = threadIdx.x & 31;
  const int wave  = threadIdx.x >> 5;
  const int which = wave >> 2;  // 0 -> W_src, 1 -> W_dst
  const int otile = wave & 3;   // 16-wide tile of D_OUT = 64
  const int row_base = blockIdx.x << 4;
  if (row_base >= BN) return;   // wave-uniform, EXEC stays all-1s for WMMA

  const int m     = lane & 15;
  const int khalf = lane >> 4;  // 0: K = 4kk+{0,1};  1: K = 4kk+{2,3}

  // Clamp row for loads so EXEC can stay full on a (hypothetical) ragged tile.
  int arow = row_base + m;
  if (arow >= BN) arow = BN - 1;

  const float* __restrict__ W = which ? Wdst : Wsrc;
  const v2f* __restrict__ xr =
      reinterpret_cast<const v2f*>(x + (size_t)arow * 64);
  const v2f* __restrict__ wr =
      reinterpret_cast<const v2f*>(W + (size_t)(otile * 16 + m) * 64);

  v8f acc = {};
#pragma unroll
  for (int kk = 0; kk < 16; ++kk) {
    const int kp = kk * 2 + khalf;  // float2 index: k = 2*kp
    const v2f a = xr[kp];
    const v2f b = wr[kp];
    // 8 args: (neg_a, A, neg_b, B, c_mod, C, reuse_a, reuse_b)
    acc = __builtin_amdgcn_wmma_f32_16x16x4_f32(false, a, false, b, (short)0,
                                                acc, false, false);
  }

  float* __restrict__ hp = h + (size_t)which * BN * 64;
  const int col = otile * 16 + m;
#pragma unroll
  for (int r = 0; r < 8; ++r) {
    const int row = row_base + (khalf ? (r + 8) : r);
    if (row < BN) hp[(size_t)row * 64 + col] = acc[r];
  }
}

// ---------------------------------------------------------------------------
// Phase 2: per-edge gather + add + scale.  16 consecutive lanes cover one
// edge's 64 output floats as 128-bit vectors (fully coalesced 256B segments).
// h reads hit L2 (h = 51.2 MB << 192 MB); output uses non-temporal stores.
// ---------------------------------------------------------------------------
__global__ void edge_gather(const float* __restrict__ h,   // [2][BN][64]
                            const int* __restrict__ src,   // [E]
                            const int* __restrict__ dst,   // [E]
                            float* __restrict__ out,       // [B][E][64]
                            int E, int N, int BN) {
  const int b = blockIdx.y;
  const long long t = (long long)blockIdx.x * blockDim.x + threadIdx.x;
  const long long e = t >> 4;
  if (e >= E) return;
  const int q = (int)(t & 15);

  const int s = src[e];
  const int d = dst[e];

  const v4f a = reinterpret_cast<const v4f*>(
      h + ((size_t)b * N + (size_t)s) * 64)[q];
  const v4f c = reinterpret_cast<const v4f*>(
      h + ((size_t)BN + (size_t)b * N + (size_t)d) * 64)[q];

  const v4f o = (a + c) * INV_SQRT_2;

  v4f* op = reinterpret_cast<v4f*>(out + ((size_t)b * E + e) * 64) + q;
  __builtin_nontemporal_store(o, op);
}

// ---------------------------------------------------------------------------
// Fallback (only if ws_size is too small for the 2*BN*64 f32 workspace):
// fuse everything per edge.  64 threads/block, one (b, e) per block.
// ---------------------------------------------------------------------------
__global__ void fused_edge_fallback(const float* __restrict__ x,
                                    const int* __restrict__ src,
                                    const int* __restrict__ dst,
                                    const float* __restrict__ Wsrc,
                                    const float* __restrict__ Wdst,
                                    float* __restrict__ out,
                                    int E, int N) {
  __shared__ float xs[64];
  __shared__ float xd[64];
  const int b = blockIdx.y;
  const int e = blockIdx.x;
  const int t = threadIdx.x;

  xs[t] = x[((size_t)b * N + (size_t)src[e]) * 64 + t];
  xd[t] = x[((size_t)b * N + (size_t)dst[e]) * 64 + t];
  __syncthreads();

  float acc = 0.0f;
#pragma unroll 8
  for (int k = 0; k < 64; ++k) {
    acc += Wsrc[t * 64 + k] * xs[k] + Wdst[t * 64 + k] * xd[k];
  }
  out[((size_t)b * E + (size_t)e) * 64 + t] = acc * INV_SQRT_2;
}

// ---------------------------------------------------------------------------
extern "C" void kernel_launch(void* const* d_in, const int* in_sizes, int n_in,
                              void* d_out, int out_size, void* d_ws,
                              size_t ws_size, hipStream_t stream) {
  const float* x    = (const float*)d_in[0];   // (B, N, 64) f32
  const int*   src  = (const int*)d_in[1];     // (E,) int
  const int*   dst  = (const int*)d_in[2];     // (E,) int
  const float* Wsrc = (const float*)d_in[3];   // (64, 64) f32
  const float* Wdst = (const float*)d_in[4];   // (64, 64) f32
  float* out = (float*)d_out;                  // (B, E, 64) f32

  const int E  = in_sizes[1];
  const int BN = in_sizes[0] / 64;        // B*N = 100000
  const int B  = out_size / (E * 64);     // 2
  const int N  = BN / B;

  const size_t need = (size_t)2 * (size_t)BN * 64 * sizeof(float);
  if (ws_size >= need) {
    float* h = (float*)d_ws;
    const int tiles = (BN + 15) / 16;
    node_transform_wmma<<<tiles, 256, 0, stream>>>(x, Wsrc, Wdst, h, BN);

    const long long threads = (long long)E * 16;
    dim3 grid((unsigned)((threads + 255) / 256), (unsigned)B);
    edge_gather<<<grid, 256, 0, stream>>>(h, src, dst, out, E, N, BN);
  } else {
    dim3 grid((unsigned)E, (unsigned)B);
    fused_edge_fallback<<<grid, 64, 0, stream>>>(x, src, dst, Wsrc, Wdst, out,
                                                 E, N);
  }
}